// GaussianRegLoss_57226144252798
// MI455X (gfx1250) — compile-verified
//
#include <hip/hip_runtime.h>
#include <hip/hip_bf16.h>

typedef __attribute__((ext_vector_type(2))) float v2f;
typedef __attribute__((ext_vector_type(8))) float v8f;

#define NJ   23
#define NVOX 262144   // 64*64*64
#define NBS  8

// ---------------------------------------------------------------------------
// init: zero the softmax-denominator workspace and the output scalar
// ---------------------------------------------------------------------------
__global__ void GaussReg_init(float* __restrict__ S, float* __restrict__ out) {
    int i = threadIdx.x;
    if (i < NBS * NJ) S[i] = 0.0f;
    if (i == 0) out[0] = 0.0f;
}

// ---------------------------------------------------------------------------
// pass 1: S[b][j] = sum_n exp( 2*exp(-d2(b,j,n)) )
// Reads grid_centers (25 MB). Scalar VALU: K=3 dot per (j,n), exp via v_exp_f32.
// gridDim.x = NBS*256, 256 threads; each thread handles 4 voxels.
// ---------------------------------------------------------------------------
__global__ void GaussReg_pass1(const float* __restrict__ kpred,
                               const float* __restrict__ grids,
                               float* __restrict__ S) {
    const int b        = blockIdx.x >> 8;
    const int tid_in_b = ((blockIdx.x & 255) << 8) | threadIdx.x;   // [0,65536)

    const float* C = kpred + b * 3 * NJ;   // [3][NJ]
    float cx[NJ], cy[NJ], cz[NJ], c2[NJ], acc[NJ];
#pragma unroll
    for (int j = 0; j < NJ; ++j) {
        cx[j] = C[j]; cy[j] = C[NJ + j]; cz[j] = C[2 * NJ + j];
        c2[j] = cx[j] * cx[j] + cy[j] * cy[j] + cz[j] * cz[j];
        acc[j] = 0.0f;
    }

    const unsigned gbase = (unsigned)b * (NVOX * 3u);
#pragma unroll
    for (int k = 0; k < 4; ++k) {
        const unsigned n = (unsigned)tid_in_b + 65536u * k;
        const unsigned go = gbase + n * 3u;
        const float gx = grids[go], gy = grids[go + 1], gz = grids[go + 2];
        const float g2 = gx * gx + gy * gy + gz * gz;
#pragma unroll
        for (int j = 0; j < NJ; ++j) {
            const float cr = gx * cx[j] + gy * cy[j] + gz * cz[j];
            const float d2 = g2 + c2[j] - 2.0f * cr;
            const float y  = 2.0f * __expf(-d2);   // exp(-d2)/(2*sigma^2), sigma=0.5
            acc[j] += __expf(y);
        }
    }

    __shared__ float red[8][NJ];
    const int lane = threadIdx.x & 31;
    const int wave = threadIdx.x >> 5;
#pragma unroll
    for (int j = 0; j < NJ; ++j) {
        float v = acc[j];
        for (int off = 16; off; off >>= 1) v += __shfl_xor(v, off, 32);
        if (lane == 0) red[wave][j] = v;
    }
    __syncthreads();
    if (threadIdx.x < NJ) {
        float s = 0.0f;
        for (int w = 0; w < 8; ++w) s += red[w][threadIdx.x];
        atomicAdd(&S[b * NJ + threadIdx.x], s);
    }
}

// ---------------------------------------------------------------------------
// pass 2: loss += sum_{j,n} ( exp(y)/S[b][j] - heat[b][j][n] )^2 / bs
// cross(j,n) via V_WMMA_F32_16X16X4_F32:  A = centers (M=j 16 x K=4),
// B = grids (K=4 x N=16 voxels), D = 16x16 j-by-n tile. Two j-tiles cover J=23.
// D layout: VGPR r, lanes 0-15 -> M=r ; lanes 16-31 -> M=r+8 ; N = lane&15.
// All heat loads are unconditional (invalid rows aliased to row 0, then
// cndmask'ed to zero) so the 193 MB stream issues as clean clauses.
// gridDim.x = NBS*128, 256 threads (8 waves); each wave loops 16 n-tiles.
// ---------------------------------------------------------------------------
__global__ void GaussReg_pass2(const float* __restrict__ kpred,
                               const float* __restrict__ heat,
                               const float* __restrict__ grids,
                               const float* __restrict__ S,
                               float* __restrict__ out) {
    const int b    = blockIdx.x >> 7;
    const int wave = threadIdx.x >> 5;
    const int lane = threadIdx.x & 31;
    const int m    = lane & 15;
    const bool hi  = lane >= 16;
    const int wave_in_b = ((blockIdx.x & 127) << 3) | wave;   // [0,1024)

    const float* C = kpred + b * 3 * NJ;

    // A operands (loop-invariant): lanes 0-15 hold K=0,1 ; lanes 16-31 hold K=2,3(=0 pad)
    v2f a[2];
#pragma unroll
    for (int t = 0; t < 2; ++t) {
        const int j = t * 16 + m;
        float a0 = 0.0f, a1 = 0.0f;
        if (j < NJ) {
            if (!hi) { a0 = C[j];          a1 = C[NJ + j]; }
            else     { a0 = C[2 * NJ + j]; a1 = 0.0f; }
        }
        a[t][0] = a0; a[t][1] = a1;
    }

    // per-(tile,r) row metadata for the j this lane-half owns in D
    float c2v[2][8], sinv[2][8];
    unsigned rowoff[2][8];      // 32-bit element offsets into heat (SADDR form)
    bool jval[2][8];
#pragma unroll
    for (int t = 0; t < 2; ++t)
#pragma unroll
        for (int r = 0; r < 8; ++r) {
            const int j  = t * 16 + (hi ? 8 : 0) + r;
            const bool v = (j < NJ);
            const int jj = v ? j : 0;            // alias invalid rows to row 0 (in-bounds)
            jval[t][r]   = v;
            rowoff[t][r] = (unsigned)(b * NJ + jj) * (unsigned)NVOX;
            const float x = C[jj], y = C[NJ + jj], z = C[2 * NJ + jj];
            c2v[t][r]  = v ? (x * x + y * y + z * z) : 0.0f;
            sinv[t][r] = v ? (1.0f / S[b * NJ + j]) : 0.0f;
        }

    const unsigned gbase = (unsigned)b * (NVOX * 3u);
    float acc = 0.0f;

#pragma unroll 2
    for (int k = 0; k < 16; ++k) {
        const int tile   = wave_in_b + 1024 * k;
        const unsigned n = (unsigned)(tile * 16 + m);

        const unsigned go = gbase + n * 3u;
        const float gx = grids[go], gy = grids[go + 1], gz = grids[go + 2];
        const float g2 = gx * gx + gy * gy + gz * gz;

        // B operand: lanes 0-15 -> K=0,1 ; lanes 16-31 -> K=2,3(=0 pad); N = lane&15
        v2f bm;
        bm[0] = hi ? gz : gx;
        bm[1] = hi ? 0.0f : gy;

#pragma unroll
        for (int t = 0; t < 2; ++t) {
            v8f d = {};
            d = __builtin_amdgcn_wmma_f32_16x16x4_f32(
                    false, a[t], false, bm, (short)0, d, false, false);
#pragma unroll
            for (int r = 0; r < 8; ++r) {
                if (t == 1 && r == 7) continue;   // j=23/31: invalid for all lanes
                // unconditional, NT-hinted, coalesced load; branchless mask after
                float h = __builtin_nontemporal_load(&heat[rowoff[t][r] + n]);
                h = jval[t][r] ? h : 0.0f;        // v_cndmask, no exec branch
                const float d2 = g2 + c2v[t][r] - 2.0f * d[r];
                const float yy = 2.0f * __expf(-d2);
                const float p  = __expf(yy) * sinv[t][r];   // invalid j -> p=0, h=0
                const float df = p - h;
                acc += df * df;
            }
        }
    }

    for (int off = 16; off; off >>= 1) acc += __shfl_xor(acc, off, 32);
    __shared__ float red[8];
    if (lane == 0) red[wave] = acc;
    __syncthreads();
    if (threadIdx.x == 0) {
        float s = 0.0f;
        for (int w = 0; w < 8; ++w) s += red[w];
        atomicAdd(out, s * (1.0f / (float)NBS));   // LOSS_WEIGHT * sum / bs
    }
}

// ---------------------------------------------------------------------------
extern "C" void kernel_launch(void* const* d_in, const int* in_sizes, int n_in,
                              void* d_out, int out_size, void* d_ws, size_t ws_size,
                              hipStream_t stream) {
    // inputs: [0] kpts_gt (unused), [1] kpts_pred, [2] heatmaps, [3] grid_centers
    const float* kpred = (const float*)d_in[1];
    const float* heat  = (const float*)d_in[2];
    const float* grids = (const float*)d_in[3];
    float* S   = (float*)d_ws;     // NBS*NJ = 184 floats
    float* out = (float*)d_out;

    GaussReg_init <<<1, 256, 0, stream>>>(S, out);
    GaussReg_pass1<<<NBS * 256, 256, 0, stream>>>(kpred, grids, S);
    GaussReg_pass2<<<NBS * 128, 256, 0, stream>>>(kpred, heat, grids, S, out);
}